// MasifLigandNet_12506944766305
// MI455X (gfx1250) — compile-verified
//
#include <hip/hip_runtime.h>
#include <hip/hip_bf16.h>

typedef __attribute__((ext_vector_type(2))) float v2f;
typedef __attribute__((ext_vector_type(8))) float v8f;

#define KNN   10
#define NPTS  8192
#define NB    32
#define NATOM 64
#define DIM   512
#define NCLS  7
#define FLT_BIG 3.0e38f

// ---------------------------------------------------------------------------
// Top-10 register insertion: sorted ascending, unrolled (compiles to cndmask
// chains, constant indices stay in VGPRs).
// ---------------------------------------------------------------------------
__device__ __forceinline__ void topk_insert(float (&bv)[KNN], int (&bi)[KNN],
                                            float c, int ci) {
    if (c < bv[KNN - 1]) {
        bv[KNN - 1] = c;
        bi[KNN - 1] = ci;
#pragma unroll
        for (int j = KNN - 1; j > 0; --j) {
            if (bv[j] < bv[j - 1]) {
                float tv = bv[j]; bv[j] = bv[j - 1]; bv[j - 1] = tv;
                int   ti = bi[j]; bi[j] = bi[j - 1]; bi[j - 1] = ti;
            }
        }
    }
}

// ---------------------------------------------------------------------------
// Kernel 1: kNN via V_WMMA_F32_16X16X4_F32.
// Block = (batch b, group g of 16 atoms). 256 threads = 8 waves.
// A row (point p):  (-2x, -2y, -2z, |p|^2)   B col (atom a): (lx, ly, lz, 1)
// C init = |l|^2 per column  =>  D = exact squared distance.
// Wave w owns point tiles w, w+8, ... (64 tiles). Each lane keeps a local
// top-10 over its 8 D rows per tile; 16 partial lists per atom merged in LDS.
// ---------------------------------------------------------------------------
__global__ __launch_bounds__(256) void knn_wmma_kernel(
    const float* __restrict__ pos, const float* __restrict__ lig,
    int* __restrict__ outIdx) {
    const int b    = blockIdx.y;
    const int g    = blockIdx.x;
    const int tid  = threadIdx.x;
    const int wave = tid >> 5;
    const int lane = tid & 31;
    const int l16  = lane & 15;
    const bool hiH = (lane >= 16);

    // --- B operand + C init (constant across tiles) ---
    const int atom = g * 16 + l16;
    const float* lp = lig + ((size_t)b * NATOM + atom) * 3;
    const float lx = lp[0], ly = lp[1], lz = lp[2];
    const float l2 = lx * lx + ly * ly + lz * lz;
    v2f bmat;
    bmat.x = hiH ? lz   : lx;   // VGPR0: rows K=0 (lo) / K=2 (hi)
    bmat.y = hiH ? 1.0f : ly;   // VGPR1: rows K=1 (lo) / K=3 (hi)
    v8f cinit;
#pragma unroll
    for (int j = 0; j < 8; ++j) cinit[j] = l2;

    float bv[KNN]; int bi[KNN];
#pragma unroll
    for (int j = 0; j < KNN; ++j) { bv[j] = FLT_BIG; bi[j] = 0; }

    const float* posb = pos + (size_t)b * NPTS * 3;
    const int rowOff = hiH ? 8 : 0;

    for (int t = wave; t < NPTS / 16; t += 8) {
        const int p0 = t * 16;
        const float* pp = posb + (size_t)(p0 + l16) * 3;
        const float px = pp[0], py = pp[1], pz = pp[2];
        const float n2 = px * px + py * py + pz * pz;
        v2f amat;
        amat.x = hiH ? (-2.0f * pz) : (-2.0f * px);  // K=2 / K=0
        amat.y = hiH ? n2           : (-2.0f * py);  // K=3 / K=1
        v8f d = __builtin_amdgcn_wmma_f32_16x16x4_f32(
            false, amat, false, bmat, (short)0, cinit, false, false);
#pragma unroll
        for (int v = 0; v < 8; ++v) {
            topk_insert(bv, bi, d[v], p0 + v + rowOff);
        }
    }

    // --- merge 16 partial lists per atom column ---
    __shared__ float mv[16][16 * KNN];
    __shared__ int   mi[16][16 * KNN];
    const int slot = wave * 2 + (hiH ? 1 : 0);
#pragma unroll
    for (int j = 0; j < KNN; ++j) {
        mv[l16][slot * KNN + j] = bv[j];
        mi[l16][slot * KNN + j] = bi[j];
    }
    __syncthreads();

    if (tid < 16) {
        float fv[KNN]; int fi[KNN];
#pragma unroll
        for (int j = 0; j < KNN; ++j) { fv[j] = FLT_BIG; fi[j] = 0; }
        for (int s = 0; s < 16 * KNN; ++s)
            topk_insert(fv, fi, mv[tid][s], mi[tid][s]);
        int* o = outIdx + ((size_t)b * NATOM + g * 16 + tid) * KNN;
#pragma unroll
        for (int j = 0; j < KNN; ++j) o[j] = fi[j];
    }
}

// ---------------------------------------------------------------------------
// Kernel 2: dedupe + masked-mean pooling. One block per batch.
// Reads only the <=640 selected feature rows (42 MB total vs 512 MB dense).
// ---------------------------------------------------------------------------
__global__ __launch_bounds__(256) void pool_kernel(
    const float* __restrict__ feats, const int* __restrict__ idxBuf,
    float* __restrict__ xOut) {
    const int b   = blockIdx.x;
    const int tid = threadIdx.x;

    __shared__ int mask[NPTS];
    __shared__ int list[NATOM * KNN];
    __shared__ int count;

    for (int i = tid; i < NPTS; i += 256) mask[i] = 0;
    if (tid == 0) count = 0;
    __syncthreads();

    const int* ib = idxBuf + (size_t)b * NATOM * KNN;
    for (int i = tid; i < NATOM * KNN; i += 256) {
        const int v = ib[i];
        if (atomicExch(&mask[v], 1) == 0) {
            const int p = atomicAdd(&count, 1);
            list[p] = v;
        }
    }
    __syncthreads();

    const int cnt = count;
    const float inv = 1.0f / (float)cnt;
    const float* fb = feats + (size_t)b * NPTS * DIM;

    float a0 = 0.0f, a1 = 0.0f;
    for (int r = 0; r < cnt; ++r) {
        const float* row = fb + (size_t)list[r] * DIM;
        if (r + 1 < cnt)  // pull next gathered row toward the caches
            __builtin_prefetch(fb + (size_t)list[r + 1] * DIM + tid * 2, 0, 0);
        const float2 v = *(const float2*)(row + tid * 2);
        a0 += v.x; a1 += v.y;
    }
    float2 res; res.x = a0 * inv; res.y = a1 * inv;
    *(float2*)(xOut + (size_t)b * DIM + tid * 2) = res;
}

// ---------------------------------------------------------------------------
// Kernel 3: classifier head, single block of 256 threads (8 waves).
// GEMM1 (32x512 @ 512x512, WMMA f32 16x16x4) + b1 -> LDS
// BatchNorm(batch stats) + SiLU in LDS -> GEMM2 (32x512 @ 512x7) scalar.
// LDS rows padded to 516 floats to de-conflict column walks.
// ---------------------------------------------------------------------------
#define LPAD 516
__global__ __launch_bounds__(256) void head_kernel(
    const float* __restrict__ xIn, const float* __restrict__ W1,
    const float* __restrict__ b1, const float* __restrict__ gamma,
    const float* __restrict__ beta, const float* __restrict__ W2,
    const float* __restrict__ b2, float* __restrict__ out) {
    __shared__ float xs[NB][LPAD];
    __shared__ float hs[NB][LPAD];

    const int tid  = threadIdx.x;
    const int wave = tid >> 5;
    const int lane = tid & 31;
    const int l16  = lane & 15;
    const bool hiH = (lane >= 16);

    for (int i = tid; i < NB * DIM; i += 256)
        xs[i / DIM][i % DIM] = xIn[i];
    __syncthreads();

    const int koff   = hiH ? 2 : 0;
    const int rowOff = hiH ? 8 : 0;

    // GEMM1: wave w owns output columns [w*64, w*64+64)
    for (int i = 0; i < 4; ++i) {
        const int nt  = wave * 4 + i;
        const int col = nt * 16 + l16;
        const float* w1r = W1 + (size_t)col * DIM;  // row of W1 == col of W1^T
        const float bb = b1[col];
        for (int mt = 0; mt < 2; ++mt) {
            const int arow = mt * 16 + l16;
            v8f acc;
#pragma unroll
            for (int j = 0; j < 8; ++j) acc[j] = 0.0f;
            for (int k = 0; k < DIM; k += 4) {
                v2f a;
                a.x = xs[arow][k + koff];
                a.y = xs[arow][k + koff + 1];
                const float2 wv = *(const float2*)(w1r + k + koff);
                v2f bm; bm.x = wv.x; bm.y = wv.y;
                acc = __builtin_amdgcn_wmma_f32_16x16x4_f32(
                    false, a, false, bm, (short)0, acc, false, false);
            }
#pragma unroll
            for (int v = 0; v < 8; ++v)
                hs[mt * 16 + v + rowOff][col] = acc[v] + bb;
        }
    }
    __syncthreads();

    // BatchNorm (batch statistics over 32 rows) + SiLU
    for (int cc = 0; cc < 2; ++cc) {
        const int c = tid * 2 + cc;
        float s = 0.0f, s2 = 0.0f;
        for (int r = 0; r < NB; ++r) {
            const float v = hs[r][c];
            s += v; s2 += v * v;
        }
        const float mu  = s * (1.0f / NB);
        const float var = s2 * (1.0f / NB) - mu * mu;
        const float rs  = rsqrtf(var + 1e-5f);
        const float ga  = gamma[c], be = beta[c];
        for (int r = 0; r < NB; ++r) {
            const float v = (hs[r][c] - mu) * rs * ga + be;
            hs[r][c] = v / (1.0f + __expf(-v));  // SiLU
        }
    }
    __syncthreads();

    // GEMM2: 32x7 outputs, one per thread
    if (tid < NB * NCLS) {
        const int r = tid / NCLS, c = tid % NCLS;
        const float* w2r = W2 + (size_t)c * DIM;
        float s = b2[c];
        for (int k = 0; k < DIM; ++k) s += hs[r][k] * w2r[k];
        out[r * NCLS + c] = s;
    }
}

// ---------------------------------------------------------------------------
// Launch: inputs in setup_inputs() order:
// 0 pos, 1 feats, 2 lig_coords, 3 W1, 4 b1, 5 gamma, 6 beta, 7 W2, 8 b2
// ---------------------------------------------------------------------------
extern "C" void kernel_launch(void* const* d_in, const int* in_sizes, int n_in,
                              void* d_out, int out_size, void* d_ws,
                              size_t ws_size, hipStream_t stream) {
    const float* pos   = (const float*)d_in[0];
    const float* feats = (const float*)d_in[1];
    const float* lig   = (const float*)d_in[2];
    const float* W1    = (const float*)d_in[3];
    const float* b1    = (const float*)d_in[4];
    const float* gamma = (const float*)d_in[5];
    const float* beta  = (const float*)d_in[6];
    const float* W2    = (const float*)d_in[7];
    const float* b2    = (const float*)d_in[8];
    float* out = (float*)d_out;

    int*   idxBuf = (int*)d_ws;                                    // 80 KB
    float* xBuf   = (float*)((char*)d_ws +
                             (size_t)NB * NATOM * KNN * sizeof(int));  // 64 KB

    knn_wmma_kernel<<<dim3(NATOM / 16, NB), 256, 0, stream>>>(pos, lig, idxBuf);
    pool_kernel<<<dim3(NB), 256, 0, stream>>>(feats, idxBuf, xBuf);
    head_kernel<<<dim3(1), 256, 0, stream>>>(xBuf, W1, b1, gamma, beta, W2, b2,
                                             out);
}